// DiffusionConvergenceDiscovery_90572270338485
// MI455X (gfx1250) — compile-verified
//
#include <hip/hip_runtime.h>

// DiffusionConvergenceDiscovery for MI455X (gfx1250, wave32).
// B=8, K=8, N=512 -> M=4096 boxes per batch. map_h=map_w=32 (512/16).
// Outputs (f32, concatenated): discovered (8,4096,4) | scores (8,4096) | heat (8,32,32)

#define BATCHES   8
#define M_BOXES   4096
#define THRESH    0.1f
#define MAP_W     32
#define MAP_H     32
#define INV2S2    0.125f   // 1/(2*2*2)

typedef __attribute__((ext_vector_type(2))) float v2f;
typedef __attribute__((ext_vector_type(8))) float v8f;

__device__ __forceinline__ bool iou_gt(float x1i, float y1i, float x2i, float y2i, float ai,
                                       const float4& bj) {
    float wj = bj.z, hj = bj.w;
    float x1j = bj.x - 0.5f * wj, y1j = bj.y - 0.5f * hj;
    float x2j = bj.x + 0.5f * wj, y2j = bj.y + 0.5f * hj;
    float ix1 = fmaxf(x1i, x1j), iy1 = fmaxf(y1i, y1j);
    float ix2 = fminf(x2i, x2j), iy2 = fminf(y2i, y2j);
    float iw = fmaxf(ix2 - ix1, 0.0f), ih = fmaxf(iy2 - iy1, 0.0f);
    float inter = iw * ih;
    float uni = ai + wj * hj - inter;
    return inter > THRESH * fmaxf(uni, 1e-6f);
}

// ---------------------------------------------------------------------------
// Kernel 1: adjacency row degree. One block = 256 rows of one batch.
// Boxes for the batch cached in 64 KB LDS (CDNA5: 320 KB per workgroup).
// ---------------------------------------------------------------------------
__global__ __launch_bounds__(256) void deg_kernel(const float* __restrict__ boxes,
                                                  int* __restrict__ deg) {
    __shared__ float4 sbox[M_BOXES];                      // 64 KB
    const int b = blockIdx.y;
    const float4* bb = (const float4*)(boxes + (size_t)b * M_BOXES * 4);
    for (int t = threadIdx.x; t < M_BOXES; t += 256) sbox[t] = bb[t];
    __syncthreads();

    const int i = blockIdx.x * 256 + threadIdx.x;
    float4 bi = sbox[i];
    float x1i = bi.x - 0.5f * bi.z, y1i = bi.y - 0.5f * bi.w;
    float x2i = bi.x + 0.5f * bi.z, y2i = bi.y + 0.5f * bi.w;
    float ai  = bi.z * bi.w;
    int d = 0;
    for (int j = 0; j < M_BOXES; ++j)
        d += iou_gt(x1i, y1i, x2i, y2i, ai, sbox[j]) ? 1 : 0;
    deg[b * M_BOXES + i] = d;
}

// ---------------------------------------------------------------------------
// Kernel 2: full clustering pipeline for one batch in one 1024-thread block.
// All state lives in LDS (224 KB < 320 KB CDNA5 workgroup LDS budget).
// ---------------------------------------------------------------------------
__global__ __launch_bounds__(1024) void cluster_kernel(const float* __restrict__ boxes,
                                                       const int* __restrict__ deg,
                                                       float* __restrict__ disc,
                                                       float* __restrict__ scores) {
    __shared__ float4 sbox[M_BOXES];        // 64 KB
    __shared__ int    skey[M_BOXES];        // 16 KB (deg, later sort key)
    __shared__ int    sorder[M_BOXES];      // 16 KB (order, later perm)
    __shared__ int    sassigned[M_BOXES];   // 16 KB
    __shared__ int    scid[M_BOXES];        // 16 KB
    __shared__ int    srank[M_BOXES];       // 16 KB
    __shared__ float  scnt[M_BOXES];        // 16 KB
    __shared__ float  scent[M_BOXES * 4];   // 64 KB

    const int b   = blockIdx.x;
    const int tid = threadIdx.x;
    const float4* bb = (const float4*)(boxes + (size_t)b * M_BOXES * 4);

    for (int i = tid; i < M_BOXES; i += 1024) {
        sbox[i]      = bb[i];
        skey[i]      = deg[b * M_BOXES + i];
        sassigned[i] = 0;
        scid[i]      = 0;
        scnt[i]      = 0.0f;
        scent[i * 4 + 0] = 0.0f; scent[i * 4 + 1] = 0.0f;
        scent[i * 4 + 2] = 0.0f; scent[i * 4 + 3] = 0.0f;
    }
    __syncthreads();

    // Stable rank == jnp.argsort(-deg): higher degree first, ties by index.
    for (int i = tid; i < M_BOXES; i += 1024) {
        int di = skey[i], p = 0;
        for (int j = 0; j < M_BOXES; ++j) {
            int dj = skey[j];
            p += (dj > di) || (dj == di && j < i);
        }
        sorder[p] = i;
    }
    __syncthreads();

    // Sequential greedy clustering scan (exact reference semantics).
    for (int t = 0; t < M_BOXES; ++t) {
        int  i   = sorder[t];
        int  a_i = sassigned[i];
        float4 bi = sbox[i];
        __syncthreads();                      // all read seed state before writes
        if (!a_i) {
            float x1i = bi.x - 0.5f * bi.z, y1i = bi.y - 0.5f * bi.w;
            float x2i = bi.x + 0.5f * bi.z, y2i = bi.y + 0.5f * bi.w;
            float ai  = bi.z * bi.w;
            for (int j = tid; j < M_BOXES; j += 1024) {
                if (!sassigned[j] && iou_gt(x1i, y1i, x2i, y2i, ai, sbox[j])) {
                    sassigned[j] = 1;         // disjoint j per thread: race-free
                    scid[j]      = i;
                }
            }
            if (tid == 0) srank[i] = t;
        } else {
            if (tid == 0) srank[i] = M_BOXES;
        }
        __syncthreads();
    }

    // Segment counts + centers via LDS float atomics (ds_add_f32).
    for (int i = tid; i < M_BOXES; i += 1024) {
        int c = scid[i];
        float4 bi = sbox[i];
        atomicAdd(&scnt[c], 1.0f);
        atomicAdd(&scent[c * 4 + 0], bi.x);
        atomicAdd(&scent[c * 4 + 1], bi.y);
        atomicAdd(&scent[c * 4 + 2], bi.z);
        atomicAdd(&scent[c * 4 + 3], bi.w);
    }
    __syncthreads();
    for (int i = tid; i < M_BOXES; i += 1024) {
        float inv = 1.0f / fmaxf(scnt[i], 1.0f);
        scent[i * 4 + 0] *= inv; scent[i * 4 + 1] *= inv;
        scent[i * 4 + 2] *= inv; scent[i * 4 + 3] *= inv;
        skey[i] = (scnt[i] >= 3.0f) ? srank[i] : M_BOXES;   // sort key #2
    }
    __syncthreads();

    // perm = stable argsort(key). Reuse sorder as perm.
    for (int i = tid; i < M_BOXES; i += 1024) {
        int ki = skey[i], p = 0;
        for (int j = 0; j < M_BOXES; ++j) {
            int kj = skey[j];
            p += (kj < ki) || (kj == ki && j < i);
        }
        sorder[p] = i;
    }
    __syncthreads();

    const float inv_m = 1.0f / (float)M_BOXES;
    for (int p = tid; p < M_BOXES; p += 1024) {
        int i = sorder[p];
        bool valid = scnt[i] >= 3.0f;
        float* dp = disc + ((size_t)b * M_BOXES + p) * 4;
        dp[0] = valid ? scent[i * 4 + 0] : 0.0f;
        dp[1] = valid ? scent[i * 4 + 1] : 0.0f;
        dp[2] = valid ? scent[i * 4 + 2] : 0.0f;
        dp[3] = valid ? scent[i * 4 + 3] : 0.0f;
        scores[b * M_BOXES + p] = valid ? scnt[i] * inv_m : 0.0f;
    }
}

// ---------------------------------------------------------------------------
// Kernel 3: heat[b] = gy^T (32x4096) x gx (4096x32) / 4096 via
// V_WMMA_F32_16X16X4_F32 (f32 in, f32 accumulate: bit-compatible precision).
// One block of 4 waves per batch; each wave owns a 16x16 output tile.
// ---------------------------------------------------------------------------
__global__ __launch_bounds__(128) void heat_kernel(const float* __restrict__ boxes,
                                                   float* __restrict__ heat) {
    __shared__ float shx[M_BOXES];   // 16 KB: clipped x bins
    __shared__ float shy[M_BOXES];   // 16 KB: clipped y bins
    const int b = blockIdx.x;
    for (int m = threadIdx.x; m < M_BOXES; m += 128) {
        float cx = boxes[((size_t)b * M_BOXES + m) * 4 + 0];
        float cy = boxes[((size_t)b * M_BOXES + m) * 4 + 1];
        shx[m] = fminf(fmaxf(truncf(cx * (float)MAP_W), 0.0f), (float)(MAP_W - 1));
        shy[m] = fminf(fmaxf(truncf(cy * (float)MAP_H), 0.0f), (float)(MAP_H - 1));
    }
    __syncthreads();

    const int wave  = threadIdx.x >> 5;
    const int lane  = threadIdx.x & 31;
    const int hbase = (wave >> 1) * 16;
    const int wbase = (wave & 1) * 16;
    const int r     = lane & 15;
    const int khi   = (lane < 16) ? 0 : 2;      // ISA A/B fragment layout
    const float fh  = (float)(hbase + r);       // A row (M) for this lane
    const float fw  = (float)(wbase + r);       // B col (N) for this lane

    v8f c = {};
    for (int mb = 0; mb < M_BOXES; mb += 4) {
        float hy0 = shy[mb + khi], hy1 = shy[mb + khi + 1];
        float hx0 = shx[mb + khi], hx1 = shx[mb + khi + 1];
        float dy0 = fh - hy0, dy1 = fh - hy1;
        float dx0 = fw - hx0, dx1 = fw - hx1;
        v2f a = { expf(-dy0 * dy0 * INV2S2), expf(-dy1 * dy1 * INV2S2) };
        v2f bm = { expf(-dx0 * dx0 * INV2S2), expf(-dx1 * dx1 * INV2S2) };
        // D = A(16x4) * B(4x16) + C  -> v_wmma_f32_16x16x4_f32
        c = __builtin_amdgcn_wmma_f32_16x16x4_f32(
                /*neg_a=*/false, a, /*neg_b=*/false, bm,
                /*c_mod=*/(short)0, c, /*reuse_a=*/false, /*reuse_b=*/false);
    }

    const float scale = 1.0f / (float)M_BOXES;
    const int row0 = hbase + ((lane < 16) ? 0 : 8);   // D layout: VGPR v -> M=v / M=v+8
    const int col  = wbase + r;
    for (int v = 0; v < 8; ++v)
        heat[b * (MAP_H * MAP_W) + (row0 + v) * MAP_W + col] = c[v] * scale;
}

// ---------------------------------------------------------------------------
extern "C" void kernel_launch(void* const* d_in, const int* in_sizes, int n_in,
                              void* d_out, int out_size, void* d_ws, size_t ws_size,
                              hipStream_t stream) {
    (void)in_sizes; (void)n_in; (void)out_size; (void)ws_size;
    const float* boxes = (const float*)d_in[0];    // (8,8,512,4) f32
    float* out = (float*)d_out;
    float* disc   = out;                                   // 8*4096*4 = 131072
    float* scores = out + BATCHES * M_BOXES * 4;           // 8*4096   = 32768
    float* heat   = scores + BATCHES * M_BOXES;            // 8*32*32  = 8192
    int*   deg    = (int*)d_ws;                            // 128 KB scratch

    deg_kernel<<<dim3(M_BOXES / 256, BATCHES), 256, 0, stream>>>(boxes, deg);
    cluster_kernel<<<BATCHES, 1024, 0, stream>>>(boxes, deg, disc, scores);
    heat_kernel<<<BATCHES, 128, 0, stream>>>(boxes, heat);
}